// MINDSSCLoss_25598005084403
// MI455X (gfx1250) — compile-verified
//
#include <hip/hip_runtime.h>

typedef float v2f __attribute__((ext_vector_type(2)));
typedef float v8f __attribute__((ext_vector_type(8)));

#define BATCH  16
#define HW     512
#define TILE   32
#define IPITCH 41   // 40-wide input tile rows, padded
#define DPITCH 37   // 36-wide d2/colsum rows, padded
#define NTILES (16 * 16 * BATCH)   // 4096 blocks per image

struct Smem {
  float in[3][40 * IPITCH];   // clamped input window, 3 channels
  float d2[36 * DPITCH];      // edge-padded diff^2 tile
  float cs[32 * DPITCH];      // vertical 5-tap sums
  float red[8];               // per-wave partials
};

// Computes the 5x5-box-filtered squared MIND-SSC difference ("ssd") for a
// 32x32 output tile; each of the 256 threads returns 4 consecutive-x values.
__device__ void compute_ssd_tile(const float* __restrict__ img, int b, int by, int bx,
                                 Smem& s, float ssd[4]) {
  const int t  = threadIdx.x;
  const int y0 = by * TILE, x0 = bx * TILE;
  const float* base = img + (size_t)b * 4 * HW * HW;

  // --- Stage clamped 40x40 window of channels 0..2 into LDS via async DMA ---
  for (int i = t; i < 3 * 40 * 40; i += 256) {
    int ch  = i / 1600;
    int rem = i - ch * 1600;
    int r = rem / 40, c = rem - r * 40;
    int gy = y0 + r - 4; gy = gy < 0 ? 0 : (gy > HW - 1 ? HW - 1 : gy);
    int gx = x0 + c - 4; gx = gx < 0 ? 0 : (gx > HW - 1 ? HW - 1 : gx);
    const float* gp = base + ((size_t)ch * HW + gy) * HW + gx;
    unsigned lds_off = (unsigned)(unsigned long long)&s.in[ch][r * IPITCH + c];
    asm volatile("global_load_async_to_lds_b32 %0, %1, off"
                 :: "v"(lds_off), "v"(gp)
                 : "memory");
  }
  asm volatile("s_wait_asynccnt 0" ::: "memory");
  __syncthreads();

  // --- diff^2 at edge-padded coordinates [-2,33]^2 (clamp the coordinate,
  //     reproducing edge_pad(diff*diff, 2)) ---
  for (int i = t; i < 36 * 36; i += 256) {
    int r = i / 36, c = i - r * 36;
    int oy = y0 + r - 2; int oyc = oy < 0 ? 0 : (oy > HW - 1 ? HW - 1 : oy);
    int ox = x0 + c - 2; int oxc = ox < 0 ? 0 : (ox > HW - 1 ? HW - 1 : ox);
    int ri = oyc - y0 + 4, ci = oxc - x0 + 4;   // tile-relative, taps +/-2 stay in [0,39]
    const float* in0 = s.in[0];
    const float* in1 = s.in[1];
    const float* in2 = s.in[2];
    float diff = in0[(ri    ) * IPITCH + ci - 2]   // ms1: ch0 (0,-2)
               + in0[(ri + 2) * IPITCH + ci + 2]   //      ch0 (2, 2)
               + in1[(ri + 2) * IPITCH + ci - 2]   //      ch1 (2,-2)
               + in2[(ri + 2) * IPITCH + ci - 2]   //      ch2 (2,-2)
               - in0[(ri - 2) * IPITCH + ci    ]   // ms2: ch0 (-2,0)
               - in1[(ri    ) * IPITCH + ci - 2]   //      ch1 (0,-2)
               - in2[(ri - 2) * IPITCH + ci    ]   //      ch2 (-2,0)
               - in2[(ri + 2) * IPITCH + ci + 2];  //      ch2 (2, 2)
    s.d2[r * DPITCH + c] = diff * diff;
  }
  __syncthreads();

  // --- separable 5x5 box: vertical 5-tap sums ---
  for (int i = t; i < 32 * 36; i += 256) {
    int y = i / 36, c = i - y * 36;
    const float* d = s.d2 + c;
    s.cs[y * DPITCH + c] = d[(y    ) * DPITCH] + d[(y + 1) * DPITCH]
                         + d[(y + 2) * DPITCH] + d[(y + 3) * DPITCH]
                         + d[(y + 4) * DPITCH];
  }
  __syncthreads();

  // --- horizontal 5-tap sums, 4 outputs per thread ---
  const int ly = t >> 3;
  const int lx = (t & 7) * 4;
  const float* cr = s.cs + ly * DPITCH + lx;
  float a0 = cr[0], a1 = cr[1], a2 = cr[2], a3 = cr[3];
  float a4 = cr[4], a5 = cr[5], a6 = cr[6], a7 = cr[7];
  const float inv25 = 1.0f / 25.0f;
  ssd[0] = (a0 + a1 + a2 + a3 + a4) * inv25;
  ssd[1] = (a1 + a2 + a3 + a4 + a5) * inv25;
  ssd[2] = (a2 + a3 + a4 + a5 + a6) * inv25;
  ssd[3] = (a3 + a4 + a5 + a6 + a7) * inv25;
  __syncthreads();   // LDS may be reused by a subsequent tile pass
}

__device__ float block_reduce_sum(float v, float* red) {
  for (int off = 16; off > 0; off >>= 1) v += __shfl_down(v, off, 32);
  int lane = threadIdx.x & 31, wv = threadIdx.x >> 5;
  if (lane == 0) red[wv] = v;
  __syncthreads();
  float r = 0.0f;
  if (wv == 0) {
    r = (lane < 8) ? red[lane] : 0.0f;
    for (int off = 4; off > 0; off >>= 1) r += __shfl_down(r, off, 32);
  }
  return r;   // valid in thread 0
}

// One wave sums n (multiple of 64) f32 partials with v_wmma_f32_16x16x4_f32:
// A = 16x4 chunk of data, B = all-ones 4x16, C accumulates.
__device__ float wave_wmma_sum(const float* __restrict__ p, int n) {
  const int lane = threadIdx.x & 31;
  const int m = lane & 15, g = lane >> 4;
  v8f acc = {0.f, 0.f, 0.f, 0.f, 0.f, 0.f, 0.f, 0.f};
  v2f bone; bone.x = 1.0f; bone.y = 1.0f;
  for (int base = 0; base < n; base += 64) {
    // A layout (32-bit 16x4): lane L, vgpr j holds A[L%16, (L/16)*2 + j]
    const float* q = p + base + m * 4 + g * 2;
    v2f a; a.x = q[0]; a.y = q[1];
    acc = __builtin_amdgcn_wmma_f32_16x16x4_f32(false, a, false, bone,
                                                (short)0, acc, false, false);
  }
  // D[m,n] identical over n; each lane's 8 components cover half of m.
  float sum = acc[0] + acc[1] + acc[2] + acc[3] + acc[4] + acc[5] + acc[6] + acc[7];
  sum += __shfl_xor(sum, 16, 32);
  return sum;
}

// Pass 1: per-tile ssd sums for both images -> partA[img*NTILES + tile]
__global__ void __launch_bounds__(256)
mind_stats_kernel(const float* __restrict__ src, const float* __restrict__ tgt,
                  float* __restrict__ partA) {
  __shared__ Smem s;
  const int z = blockIdx.z;            // 0..31: img*16 + b
  const int img = z >> 4, b = z & 15;
  const float* im = img ? tgt : src;
  float ssd[4];
  compute_ssd_tile(im, b, blockIdx.y, blockIdx.x, s, ssd);
  float tot = block_reduce_sum(ssd[0] + ssd[1] + ssd[2] + ssd[3], s.red);
  if (threadIdx.x == 0)
    partA[img * NTILES + ((b * 16 + blockIdx.y) * 16 + blockIdx.x)] = tot;
}

// Pass 2: mv[img] = mean(ssd)/4 via WMMA reduction of 4096 partials
__global__ void mv_reduce_kernel(const float* __restrict__ partA, float* __restrict__ mv) {
  const int img = blockIdx.x;
  float tot = wave_wmma_sum(partA + img * NTILES, NTILES);
  if (threadIdx.x == 0)
    mv[img] = tot * (0.25f / ((float)BATCH * HW * HW));
}

// Pass 3: recompute ssd for both images per tile, apply clip/exp, reduce (e_s-e_t)^2
__global__ void __launch_bounds__(256)
mind_loss_kernel(const float* __restrict__ src, const float* __restrict__ tgt,
                 const float* __restrict__ mv, float* __restrict__ partL) {
  __shared__ Smem s;
  const int b = blockIdx.z;
  float ssdS[4], ssdT[4];
  compute_ssd_tile(src, b, blockIdx.y, blockIdx.x, s, ssdS);
  compute_ssd_tile(tgt, b, blockIdx.y, blockIdx.x, s, ssdT);
  const float mvS = mv[0], mvT = mv[1];
  float acc = 0.0f;
  #pragma unroll
  for (int i = 0; i < 4; ++i) {
    float vs = fminf(fmaxf(ssdS[i] * 0.25f, mvS * 1e-3f), mvS * 1e3f);
    float vt = fminf(fmaxf(ssdT[i] * 0.25f, mvT * 1e-3f), mvT * 1e3f);
    float d  = __expf(-ssdS[i] / vs) - __expf(-ssdT[i] / vt);
    acc += d * d;
  }
  float tot = block_reduce_sum(acc, s.red);
  if (threadIdx.x == 0)
    partL[(b * 16 + blockIdx.y) * 16 + blockIdx.x] = tot;
}

// Pass 4: final scalar = sum(partL) / (B*3*H*W)
__global__ void loss_final_kernel(const float* __restrict__ partL, float* __restrict__ out) {
  float tot = wave_wmma_sum(partL, NTILES);
  if (threadIdx.x == 0)
    out[0] = tot / ((float)BATCH * 3.0f * HW * HW);
}

extern "C" void kernel_launch(void* const* d_in, const int* in_sizes, int n_in,
                              void* d_out, int out_size, void* d_ws, size_t ws_size,
                              hipStream_t stream) {
  const float* src = (const float*)d_in[0];
  const float* tgt = (const float*)d_in[1];
  float* out = (float*)d_out;

  float* partA = (float*)d_ws;           // 2 * 4096 floats
  float* mvbuf = partA + 2 * NTILES;     // 2 floats
  float* partL = mvbuf + 2;              // 4096 floats   (~50 KB total)

  dim3 blk(256);
  mind_stats_kernel<<<dim3(16, 16, 32), blk, 0, stream>>>(src, tgt, partA);
  mv_reduce_kernel<<<dim3(2), dim3(32), 0, stream>>>(partA, mvbuf);
  mind_loss_kernel<<<dim3(16, 16, BATCH), blk, 0, stream>>>(src, tgt, mvbuf, partL);
  loss_final_kernel<<<dim3(1), dim3(32), 0, stream>>>(partL, out);
}